// OTAlignmentLoss_56719338111040
// MI455X (gfx1250) — compile-verified
//
#include <hip/hip_runtime.h>
#include <math.h>

#define REGC    0.05f
#define INV_REG 20.0f
#define EPSC    1e-8f
#define NPTS    4096
#define DIM     128
#define N_ITER  100

typedef __attribute__((ext_vector_type(2))) float v2f;
typedef __attribute__((ext_vector_type(8))) float v8f;

// ---------------------------------------------------------------- small utils
__global__ void zero_out_kernel(float* __restrict__ out) { out[0] = 0.0f; }

__global__ void init_ones_kernel(float* __restrict__ u) {
  u[blockIdx.x * blockDim.x + threadIdx.x] = 1.0f;
}

// per-row sum of squares: 4096 threads, one row each (32 float4 loads)
__global__ __launch_bounds__(256) void rowsq_kernel(const float* __restrict__ X,
                                                    float* __restrict__ sq) {
  const int row = blockIdx.x * blockDim.x + threadIdx.x;
  const float4* xr = (const float4*)(X + (size_t)row * DIM);
  float acc = 0.0f;
#pragma unroll
  for (int i = 0; i < DIM / 4; ++i) {
    float4 t = xr[i];
    acc += t.x * t.x + t.y * t.y + t.z * t.z + t.w * t.w;
  }
  sq[row] = acc;
}

// ---------------------------------------------------------------- Gram + exp
// Each wave computes one 16x16 tile of G = X * Y^T with f32 WMMA (K-step 4),
// then K = exp(-clip(xsq+ysq-2G,0)/REG), stored to both K and K^T buffers.
//
// A (16x4 f32) lane layout: M = lane%16; VGPR0/1 hold K = 2*(lane>>4) + {0,1}.
// B (4x16 f32) mirrors on Y rows (N = lane%16).
// C/D (16x16 f32): VGPR r -> M = r + 8*(lane>>4), N = lane%16.
__global__ __launch_bounds__(256) void gram_exp_kernel(
    const float* __restrict__ X, const float* __restrict__ Y,
    const float* __restrict__ xsq, const float* __restrict__ ysq,
    float* __restrict__ K, float* __restrict__ KT) {
  const int lane = threadIdx.x & 31;
  const int wave = threadIdx.x >> 5;
  const int tile = blockIdx.x * 8 + wave;   // 0 .. 65535
  const int tm   = (tile >> 8) << 4;        // tile row base
  const int tn   = (tile & 255) << 4;       // tile col base
  const int lo   = lane & 15;
  const int hi   = lane >> 4;

  const float* arow = X + (size_t)(tm + lo) * DIM + 2 * hi;
  const float* brow = Y + (size_t)(tn + lo) * DIM + 2 * hi;

  v8f c = {};
#pragma unroll
  for (int k0 = 0; k0 < DIM; k0 += 4) {
    v2f a = *(const v2f*)(arow + k0);
    v2f b = *(const v2f*)(brow + k0);
    c = __builtin_amdgcn_wmma_f32_16x16x4_f32(false, a, false, b,
                                              (short)0, c, false, false);
  }

  const int   n     = tn + lo;
  const float ysq_n = ysq[n];
#pragma unroll
  for (int r = 0; r < 8; ++r) {
    const int   m    = tm + r + 8 * hi;
    const float g    = c[r];
    const float cost = fmaxf(xsq[m] + ysq_n - 2.0f * g, 0.0f);
    const float kv   = expf(-cost * INV_REG);
    K [(size_t)m * NPTS + n] = kv;
    KT[(size_t)n * NPTS + m] = kv;
  }
}

// ---------------------------------------------------------------- Sinkhorn step
// out[row] = scale / (dot(M[row,:], w) + eps).  Wave per row, coalesced float4.
__global__ __launch_bounds__(256) void step_kernel(const float* __restrict__ M,
                                                   const float* __restrict__ w,
                                                   float* __restrict__ out) {
  const int lane = threadIdx.x & 31;
  const int wave = threadIdx.x >> 5;
  const int row  = blockIdx.x * 8 + wave;
  const float4* mrow = (const float4*)(M + (size_t)row * NPTS);
  const float4* wv   = (const float4*)w;
  float acc = 0.0f;
#pragma unroll 4
  for (int j = lane; j < NPTS / 4; j += 32) {
    float4 a = mrow[j];
    float4 b = wv[j];
    acc += a.x * b.x + a.y * b.y + a.z * b.z + a.w * b.w;
  }
#pragma unroll
  for (int off = 16; off > 0; off >>= 1) acc += __shfl_xor(acc, off, 32);
  if (lane == 0) out[row] = (1.0f / (float)NPTS) / (acc + EPSC);
}

// ---------------------------------------------------------------- final loss
// sum_ij u_i * K_ij * v_j * C_ij with C_ij = -REG*log(K_ij); scaled by 1/3.
__global__ __launch_bounds__(256) void loss_kernel(const float* __restrict__ K,
                                                   const float* __restrict__ u,
                                                   const float* __restrict__ v,
                                                   float* __restrict__ out) {
  const int lane = threadIdx.x & 31;
  const int wave = threadIdx.x >> 5;
  const int row  = blockIdx.x * 8 + wave;
  const float4* krow = (const float4*)(K + (size_t)row * NPTS);
  const float4* vv   = (const float4*)v;
  float acc = 0.0f;
#pragma unroll 2
  for (int j = lane; j < NPTS / 4; j += 32) {
    float4 a = krow[j];
    float4 b = vv[j];
    float c0 = (a.x > 0.0f) ? (-REGC * logf(a.x)) : 0.0f;
    float c1 = (a.y > 0.0f) ? (-REGC * logf(a.y)) : 0.0f;
    float c2 = (a.z > 0.0f) ? (-REGC * logf(a.z)) : 0.0f;
    float c3 = (a.w > 0.0f) ? (-REGC * logf(a.w)) : 0.0f;
    acc += a.x * b.x * c0 + a.y * b.y * c1 + a.z * b.z * c2 + a.w * b.w * c3;
  }
#pragma unroll
  for (int off = 16; off > 0; off >>= 1) acc += __shfl_xor(acc, off, 32);
  if (lane == 0) atomicAdd(out, u[row] * acc * (1.0f / 3.0f));
}

// ---------------------------------------------------------------- launch
extern "C" void kernel_launch(void* const* d_in, const int* in_sizes, int n_in,
                              void* d_out, int out_size, void* d_ws, size_t ws_size,
                              hipStream_t stream) {
  (void)in_sizes; (void)n_in; (void)out_size; (void)ws_size;
  const float* z0 = (const float*)d_in[0];
  const float* z1 = (const float*)d_in[1];
  const float* z2 = (const float*)d_in[2];
  float* out = (float*)d_out;

  float* ws  = (float*)d_ws;
  float* K   = ws;                                  // 4096*4096 f32 (64 MB)
  float* KT  = K  + (size_t)NPTS * NPTS;            // 4096*4096 f32 (64 MB)
  float* xsq = KT + (size_t)NPTS * NPTS;            // 4096
  float* ysq = xsq + NPTS;                          // 4096
  float* u   = ysq + NPTS;                          // 4096
  float* v   = u   + NPTS;                          // 4096

  zero_out_kernel<<<1, 1, 0, stream>>>(out);

  const float* pairs[3][2] = {{z0, z1}, {z0, z2}, {z1, z2}};
  for (int p = 0; p < 3; ++p) {
    const float* X = pairs[p][0];
    const float* Y = pairs[p][1];

    rowsq_kernel<<<NPTS / 256, 256, 0, stream>>>(X, xsq);
    rowsq_kernel<<<NPTS / 256, 256, 0, stream>>>(Y, ysq);
    gram_exp_kernel<<<(NPTS / 16) * (NPTS / 16) / 8, 256, 0, stream>>>(
        X, Y, xsq, ysq, K, KT);
    init_ones_kernel<<<NPTS / 256, 256, 0, stream>>>(u);

    for (int it = 0; it < N_ITER; ++it) {
      step_kernel<<<NPTS / 8, 256, 0, stream>>>(KT, u, v);  // v = nu/(K^T u + eps)
      step_kernel<<<NPTS / 8, 256, 0, stream>>>(K,  v, u);  // u = mu/(K v + eps)
    }
    loss_kernel<<<NPTS / 8, 256, 0, stream>>>(K, u, v, out);
  }
}